// EnsembleMetaLearner_47605417508994
// MI455X (gfx1250) — compile-verified
//
#include <hip/hip_runtime.h>
#include <math.h>

// ---------------------------------------------------------------------------
// CDNA5 fp32 WMMA types
// ---------------------------------------------------------------------------
typedef __attribute__((ext_vector_type(2))) float v2f;
typedef __attribute__((ext_vector_type(8))) float v8f;

#define NW     4      // waves per block (128 threads, wave32)
#define LD256  260    // padded LDS row stride for 256-wide tiles (bank-safe)
#define LD400  404    // padded LDS row stride for 400-wide tiles
#define B_     1024
#define M_     400
#define H_     256
#define D_     3

__device__ __forceinline__ float gelu_exact(float x) {
  return 0.5f * x * (1.0f + erff(x * 0.70710678118654752f));
}

// ---------------------------------------------------------------------------
// O[16 x ncols] (LDS) = scale * (A[16 x K] (LDS) @ W[K x *] (global)) + bias
// A row stride = lda (padded), W row stride = ldw, O row stride = ldo.
// Uses V_WMMA_F32_16X16X4_F32; waves split the 16-wide column tiles.
//
// Fragment layouts per CDNA5 ISA 7.12.2:
//   A 16x4 : lane holds row (lane&15); VGPR0/1 = K+{0,1} (lanes 0-15)
//            or K+{2,3} (lanes 16-31)
//   B 4x16 : VGPR0 = row K+0 (lanes0-15)/K+2 (lanes16-31), VGPR1 = K+1/K+3
//   D 16x16: VGPR r = row r (lanes0-15) / row r+8 (lanes16-31), col = lane&15
// ---------------------------------------------------------------------------
__device__ __forceinline__ void gemm16(const float* __restrict__ A, int lda,
                                       const float* __restrict__ W, int ldw,
                                       const float* __restrict__ bias,
                                       float* __restrict__ O, int ldo,
                                       int K, int ncols, float scale, int tid)
{
  const int wave = tid >> 5;
  const int lane = tid & 31;
  const int half = lane >> 4;     // 0 or 1
  const int l16  = lane & 15;
  const int koff = half << 1;     // 0 or 2
  for (int nt = wave; nt < (ncols >> 4); nt += NW) {
    const int col = (nt << 4) + l16;
    v8f c = {0.f, 0.f, 0.f, 0.f, 0.f, 0.f, 0.f, 0.f};
    const float* arow = A + l16 * lda;
    for (int k0 = 0; k0 < K; k0 += 4) {
      v2f a, b;
      a.x = arow[k0 + koff];
      a.y = arow[k0 + koff + 1];
      b.x = W[(k0 + koff)     * ldw + col];
      b.y = W[(k0 + koff + 1) * ldw + col];
      c = __builtin_amdgcn_wmma_f32_16x16x4_f32(false, a, false, b,
                                                (short)0, c, false, false);
    }
    const float bb = bias ? bias[col] : 0.f;
#pragma unroll
    for (int r = 0; r < 8; ++r) {
      O[(r + half * 8) * ldo + col] = c[r] * scale + bb;
    }
  }
}

// LayerNorm over 256 cols then exact GELU, per row; 8 lanes per row.
__device__ __forceinline__ void ln_gelu_rows(float* buf, int ld,
                                             const float* g, const float* b,
                                             int tid)
{
  const int row = tid >> 3, sub = tid & 7;
  float s = 0.f;
  for (int c = sub; c < 256; c += 8) s += buf[row * ld + c];
#pragma unroll
  for (int m = 1; m < 8; m <<= 1) s += __shfl_xor(s, m, 8);
  const float mean = s * (1.f / 256.f);
  float v = 0.f;
  for (int c = sub; c < 256; c += 8) {
    float d = buf[row * ld + c] - mean; v += d * d;
  }
#pragma unroll
  for (int m = 1; m < 8; m <<= 1) v += __shfl_xor(v, m, 8);
  const float inv = rsqrtf(v * (1.f / 256.f) + 1e-5f);
  for (int c = sub; c < 256; c += 8) {
    float x = (buf[row * ld + c] - mean) * inv * g[c] + b[c];
    buf[row * ld + c] = gelu_exact(x);
  }
}

__device__ __forceinline__ void gelu_inplace(float* buf, int ld, int tid) {
  for (int i = tid; i < 16 * 256; i += 128) {
    int r = i >> 8, c = i & 255;
    buf[r * ld + c] = gelu_exact(buf[r * ld + c]);
  }
}

// softmax over 400 cols per row; 8 lanes per row
__device__ __forceinline__ void softmax_rows400(float* buf, int ld, int tid) {
  const int row = tid >> 3, sub = tid & 7;
  float mx = -3.4e38f;
  for (int m = sub; m < 400; m += 8) mx = fmaxf(mx, buf[row * ld + m]);
#pragma unroll
  for (int k = 1; k < 8; k <<= 1) mx = fmaxf(mx, __shfl_xor(mx, k, 8));
  float s = 0.f;
  for (int m = sub; m < 400; m += 8) {
    float e = expf(buf[row * ld + m] - mx);
    buf[row * ld + m] = e; s += e;
  }
#pragma unroll
  for (int k = 1; k < 8; k <<= 1) s += __shfl_xor(s, k, 8);
  const float inv = 1.f / s;
  for (int m = sub; m < 400; m += 8) buf[row * ld + m] *= inv;
}

// ---------------------------------------------------------------------------
// Kernel A: B-independent member precompute.
//   me = member_emb @ proj_w + proj_b           -> stored transposed meT[256][400]
//   k  = me @ attn_wk + bk                      -> stored transposed kT [256][400]
//   v  = me @ attn_wv + bv                      -> stored row-major   v  [400][256]
// One 16-member tile per workgroup (25 blocks).
// ---------------------------------------------------------------------------
struct PreParams {
  const float *member_emb, *proj_w, *proj_b;
  const float *attn_wk, *attn_bk, *attn_wv, *attn_bv;
  float *meT, *kT, *v;
};

__global__ void __launch_bounds__(128) precompute_members(PreParams p) {
  __shared__ float emb[16 * 68];
  __shared__ float me [16 * LD256];
  __shared__ float tmp[16 * LD256];
  const int tid = threadIdx.x;
  const int m0  = blockIdx.x * 16;

  for (int i = tid; i < 16 * 64; i += 128) {
    int r = i >> 6, c = i & 63;
    emb[r * 68 + c] = p.member_emb[(m0 + r) * 64 + c];
  }
  __syncthreads();
  gemm16(emb, 68, p.proj_w, 256, p.proj_b, me, LD256, 64, 256, 1.f, tid);
  __syncthreads();
  for (int i = tid; i < 16 * 256; i += 128) {       // write meT (transposed)
    int d = i >> 4, r = i & 15;
    p.meT[d * 400 + m0 + r] = me[r * LD256 + d];
  }
  gemm16(me, LD256, p.attn_wk, 256, p.attn_bk, tmp, LD256, 256, 256, 1.f, tid);
  __syncthreads();
  for (int i = tid; i < 16 * 256; i += 128) {       // write kT (transposed)
    int d = i >> 4, r = i & 15;
    p.kT[d * 400 + m0 + r] = tmp[r * LD256 + d];
  }
  __syncthreads();
  gemm16(me, LD256, p.attn_wv, 256, p.attn_bv, tmp, LD256, 256, 256, 1.f, tid);
  __syncthreads();
  for (int i = tid; i < 16 * 256; i += 128) {       // write v (row-major)
    int r = i >> 8, c = i & 255;
    p.v[(m0 + r) * 256 + c] = tmp[r * LD256 + c];
  }
}

// ---------------------------------------------------------------------------
// Kernel B: fused per-sample chain, 16 rows of B per workgroup (64 blocks).
// encoder -> hypernet base weights -> 4-head single-query attention ->
// combined softmax/clip/renorm -> prediction.
// ---------------------------------------------------------------------------
struct FuseParams {
  const float *market, *preds, *regime_tab;
  const float *enc_w1, *enc_b1, *ln1_g, *ln1_b;
  const float *enc_w2, *enc_b2, *ln2_g, *ln2_b;
  const float *enc_w3, *enc_b3;
  const float *wg_w1, *wg_b1, *wg_w2, *wg_b2, *wg_w3, *wg_b3;
  const float *wq, *bq, *wo, *bo;
  const float *meT, *kT, *v;
  const int   *rid;
  float *pred_out, *w_out;
};

__global__ void __launch_bounds__(128) fused_ensemble(FuseParams p) {
  __shared__ float ping[16 * LD256];
  __shared__ float pong[16 * LD256];
  __shared__ float ctx [16 * LD256];
  __shared__ float bw  [16 * LD400];   // base weights -> combined -> final w
  __shared__ float sc  [16 * LD400];   // per-head scores, then attention_weights
  const int tid = threadIdx.x;
  const int b0  = blockIdx.x * 16;

  // ---- concat(market, regime_emb[rid]) -> ping[16][160] ----
  for (int i = tid; i < 16 * 160; i += 128) {
    int r = i / 160, c = i - r * 160;
    float val = (c < 128) ? p.market[(b0 + r) * 128 + c]
                          : p.regime_tab[p.rid[b0 + r] * 32 + (c - 128)];
    ping[r * LD256 + c] = val;
  }
  __syncthreads();

  // ---- ContextEncoder ----
  gemm16(ping, LD256, p.enc_w1, 256, p.enc_b1, pong, LD256, 160, 256, 1.f, tid);
  __syncthreads();
  ln_gelu_rows(pong, LD256, p.ln1_g, p.ln1_b, tid);
  __syncthreads();
  gemm16(pong, LD256, p.enc_w2, 256, p.enc_b2, ping, LD256, 256, 256, 1.f, tid);
  __syncthreads();
  ln_gelu_rows(ping, LD256, p.ln2_g, p.ln2_b, tid);
  __syncthreads();
  gemm16(ping, LD256, p.enc_w3, 256, p.enc_b3, ctx, LD256, 256, 256, 1.f, tid);
  __syncthreads();

  // ---- HyperNetwork base weights ----
  gemm16(ctx, LD256, p.wg_w1, 256, p.wg_b1, ping, LD256, 256, 256, 1.f, tid);
  __syncthreads();
  gelu_inplace(ping, LD256, tid);
  __syncthreads();
  gemm16(ping, LD256, p.wg_w2, 256, p.wg_b2, pong, LD256, 256, 256, 1.f, tid);
  __syncthreads();
  gelu_inplace(pong, LD256, tid);
  __syncthreads();
  gemm16(pong, LD256, p.wg_w3, 400, p.wg_b3, bw, LD400, 256, 400, 1.f, tid);
  __syncthreads();

  // ---- attention: q from ctx ----
  gemm16(ctx, LD256, p.wq, 256, p.bq, ping, LD256, 256, 256, 1.f, tid);
  __syncthreads();

  // per head: scores = q_h @ k_h^T / 8 -> softmax -> att_h = a @ v_h
  for (int h = 0; h < 4; ++h) {
    gemm16(&ping[h * 64], LD256, p.kT + h * 64 * 400, 400, nullptr,
           sc, LD400, 64, 400, 0.125f, tid);
    __syncthreads();
    softmax_rows400(sc, LD400, tid);
    __syncthreads();
    gemm16(sc, LD400, p.v + h * 64, 256, nullptr,
           &pong[h * 64], LD256, 400, 64, 1.f, tid);
    __syncthreads();
  }

  // attended = att @ wo + bo
  gemm16(pong, LD256, p.wo, 256, p.bo, ping, LD256, 256, 256, 1.f, tid);
  __syncthreads();
  // attention_weights = attended @ me^T   (reuse sc buffer)
  gemm16(ping, LD256, p.meT, 400, nullptr, sc, LD400, 256, 400, 1.f, tid);
  __syncthreads();

  // ---- combined -> softmax -> clip(0.001) -> renorm -> w ----
  {
    const int row = tid >> 3, sub = tid & 7;
    float mx = -3.4e38f;
    for (int m = sub; m < 400; m += 8) {
      float c = bw[row * LD400 + m] + 0.5f * sc[row * LD400 + m];
      bw[row * LD400 + m] = c;
      mx = fmaxf(mx, c);
    }
#pragma unroll
    for (int k = 1; k < 8; k <<= 1) mx = fmaxf(mx, __shfl_xor(mx, k, 8));
    float s = 0.f;
    for (int m = sub; m < 400; m += 8) {
      float e = expf(bw[row * LD400 + m] - mx);
      bw[row * LD400 + m] = e; s += e;
    }
#pragma unroll
    for (int k = 1; k < 8; k <<= 1) s += __shfl_xor(s, k, 8);
    const float inv = 1.f / s;
    float s2 = 0.f;
    for (int m = sub; m < 400; m += 8) {
      float wv = fmaxf(bw[row * LD400 + m] * inv, 0.001f);
      bw[row * LD400 + m] = wv; s2 += wv;
    }
#pragma unroll
    for (int k = 1; k < 8; k <<= 1) s2 += __shfl_xor(s2, k, 8);
    const float inv2 = 1.f / s2;
    for (int m = sub; m < 400; m += 8) {
      float wv = bw[row * LD400 + m] * inv2;
      bw[row * LD400 + m] = wv;
      p.w_out[(b0 + row) * 400 + m] = wv;
    }
  }
  __syncthreads();

  // ---- prediction = w @ member_predictions (D=3) ----
  if (tid < 48) {
    const int r = tid / 3, d = tid - r * 3;
    const float* pr = p.preds + (size_t)(b0 + r) * (M_ * D_) + d;
    float acc = 0.f;
    for (int m = 0; m < 400; ++m)
      acc += bw[r * LD400 + m] * pr[m * 3];
    p.pred_out[(b0 + r) * 3 + d] = acc;
  }
}

// ---------------------------------------------------------------------------
// Diversity: offdiag_sum = sum_b ( ||sum_m pn||^2 - sum_m ||pn_m||^2 )
// One wave per sample; 32 block partials then a final reduce.
// ---------------------------------------------------------------------------
__global__ void __launch_bounds__(256) diversity_partial(const float* __restrict__ preds,
                                                         float* __restrict__ partials) {
  const int wave = threadIdx.x >> 5, lane = threadIdx.x & 31;
  float acc = 0.f;
  for (int b = blockIdx.x * 8 + wave; b < B_; b += gridDim.x * 8) {
    float sx = 0.f, sy = 0.f, sz = 0.f, t = 0.f;
    for (int m = lane; m < M_; m += 32) {
      const float* q = preds + ((size_t)b * M_ + m) * 3;
      float x = q[0], y = q[1], z = q[2];
      float n = fmaxf(sqrtf(x * x + y * y + z * z), 1e-12f);
      float inv = 1.f / n;
      x *= inv; y *= inv; z *= inv;
      sx += x; sy += y; sz += z;
      t += x * x + y * y + z * z;
    }
#pragma unroll
    for (int k = 1; k < 32; k <<= 1) {
      sx += __shfl_xor(sx, k); sy += __shfl_xor(sy, k);
      sz += __shfl_xor(sz, k); t  += __shfl_xor(t, k);
    }
    if (lane == 0) acc += sx * sx + sy * sy + sz * sz - t;
  }
  __shared__ float red[8];
  if (lane == 0) red[wave] = acc;
  __syncthreads();
  if (threadIdx.x == 0) {
    float s = 0.f;
    for (int i = 0; i < 8; ++i) s += red[i];
    partials[blockIdx.x] = s;
  }
}

__global__ void diversity_final(const float* __restrict__ partials, float* __restrict__ out) {
  if (threadIdx.x == 0) {
    float s = 0.f;
    for (int i = 0; i < 32; ++i) s += partials[i];
    out[0] = s / ((float)B_ * (float)M_ * (float)(M_ - 1)) * 0.1f;
  }
}

// ---------------------------------------------------------------------------
extern "C" void kernel_launch(void* const* d_in, const int* in_sizes, int n_in,
                              void* d_out, int out_size, void* d_ws, size_t ws_size,
                              hipStream_t stream) {
  (void)in_sizes; (void)n_in; (void)out_size; (void)ws_size;

  const float* market      = (const float*)d_in[0];
  const float* preds       = (const float*)d_in[1];
  const float* regime_tab  = (const float*)d_in[2];
  const float* enc_w1      = (const float*)d_in[3];
  const float* enc_b1      = (const float*)d_in[4];
  const float* ln1_g       = (const float*)d_in[5];
  const float* ln1_b       = (const float*)d_in[6];
  const float* enc_w2      = (const float*)d_in[7];
  const float* enc_b2      = (const float*)d_in[8];
  const float* ln2_g       = (const float*)d_in[9];
  const float* ln2_b       = (const float*)d_in[10];
  const float* enc_w3      = (const float*)d_in[11];
  const float* enc_b3      = (const float*)d_in[12];
  const float* wg_w1       = (const float*)d_in[13];
  const float* wg_b1       = (const float*)d_in[14];
  const float* wg_w2       = (const float*)d_in[15];
  const float* wg_b2       = (const float*)d_in[16];
  const float* wg_w3       = (const float*)d_in[17];
  const float* wg_b3       = (const float*)d_in[18];
  const float* member_emb  = (const float*)d_in[19];
  const float* proj_w      = (const float*)d_in[20];
  const float* proj_b      = (const float*)d_in[21];
  const float* attn_wq     = (const float*)d_in[22];
  const float* attn_bq     = (const float*)d_in[23];
  const float* attn_wk     = (const float*)d_in[24];
  const float* attn_bk     = (const float*)d_in[25];
  const float* attn_wv     = (const float*)d_in[26];
  const float* attn_bv     = (const float*)d_in[27];
  const float* attn_wo     = (const float*)d_in[28];
  const float* attn_bo     = (const float*)d_in[29];
  const int*   rid         = (const int*)d_in[30];

  float* out      = (float*)d_out;
  float* pred_out = out;                       // [1024, 3]
  float* w_out    = out + B_ * 3;              // [1024, 400]
  float* div_out  = out + B_ * 3 + B_ * M_;    // scalar

  float* ws    = (float*)d_ws;
  float* meT   = ws;                  // [256][400]
  float* kT    = ws + 102400;         // [256][400]
  float* vbuf  = ws + 204800;         // [400][256]
  float* parts = ws + 307200;         // [32]

  PreParams pp = { member_emb, proj_w, proj_b,
                   attn_wk, attn_bk, attn_wv, attn_bv,
                   meT, kT, vbuf };
  precompute_members<<<25, 128, 0, stream>>>(pp);

  FuseParams fp;
  fp.market = market;   fp.preds = preds;   fp.regime_tab = regime_tab;
  fp.enc_w1 = enc_w1;   fp.enc_b1 = enc_b1; fp.ln1_g = ln1_g; fp.ln1_b = ln1_b;
  fp.enc_w2 = enc_w2;   fp.enc_b2 = enc_b2; fp.ln2_g = ln2_g; fp.ln2_b = ln2_b;
  fp.enc_w3 = enc_w3;   fp.enc_b3 = enc_b3;
  fp.wg_w1 = wg_w1;     fp.wg_b1 = wg_b1;
  fp.wg_w2 = wg_w2;     fp.wg_b2 = wg_b2;
  fp.wg_w3 = wg_w3;     fp.wg_b3 = wg_b3;
  fp.wq = attn_wq;      fp.bq = attn_bq;
  fp.wo = attn_wo;      fp.bo = attn_bo;
  fp.meT = meT;         fp.kT = kT;         fp.v = vbuf;
  fp.rid = rid;
  fp.pred_out = pred_out; fp.w_out = w_out;
  fused_ensemble<<<64, 128, 0, stream>>>(fp);

  diversity_partial<<<32, 256, 0, stream>>>(preds, parts);
  diversity_final<<<1, 32, 0, stream>>>(parts, div_out);
}